// Encoder_16595753632514
// MI455X (gfx1250) — compile-verified
//
#include <hip/hip_runtime.h>

typedef __attribute__((ext_vector_type(2))) float v2f;
typedef __attribute__((ext_vector_type(8))) float v8f;

#define N_NODES 50000
#define N_EDGES 800000
#define D 128
#define NB (N_NODES * D)

// ---------------------------------------------------------------------------
// Zero agg[NB] and deg[N_NODES]
// ---------------------------------------------------------------------------
__global__ void zero_kernel(float* __restrict__ agg, float* __restrict__ deg) {
    int i = blockIdx.x * blockDim.x + threadIdx.x;
    if (i < NB) agg[i] = 0.0f;
    if (i < N_NODES) deg[i] = 0.0f;
}

// ---------------------------------------------------------------------------
// Edge scatter: one wave owns 32 edges. Each lane loads one edge's (src,dst)
// pair with coalesced b64 loads, then the wave iterates over the 32 edges,
// broadcasting src/dst via __shfl (v_readlane -> SGPR). For each edge all 32
// lanes move 4 contiguous floats (float4 load -> 4 global fp32 atomics).
// Working set (x, agg) is L2-resident, so atomics resolve at L2 rate.
// Degree: one coalesced atomic per lane after the loop (one per edge).
// ---------------------------------------------------------------------------
__global__ void scatter_kernel(const float* __restrict__ x,
                               const long long* __restrict__ ei,
                               float* __restrict__ agg,
                               float* __restrict__ deg) {
    const int gtid = blockIdx.x * blockDim.x + threadIdx.x;
    const int wave = gtid >> 5;
    const int lane = threadIdx.x & 31;
    const int e0 = wave * 32;
    if (e0 >= N_EDGES) return;

    const int mysrc = (int)ei[e0 + lane];            // coalesced b64
    const int mydst = (int)ei[N_EDGES + e0 + lane];  // coalesced b64

    #pragma unroll 4
    for (int i = 0; i < 32; ++i) {
        const int src = __shfl(mysrc, i, 32);        // wave-uniform broadcast
        const int dst = __shfl(mydst, i, 32);
        const float4 v =
            *reinterpret_cast<const float4*>(x + (size_t)src * D + lane * 4);
        float* base = agg + (size_t)dst * D + lane * 4;
        atomicAdd(base + 0, v.x);
        atomicAdd(base + 1, v.y);
        atomicAdd(base + 2, v.z);
        atomicAdd(base + 3, v.w);
    }
    atomicAdd(deg + mydst, 1.0f);                    // one per edge, per lane
}

// ---------------------------------------------------------------------------
// Skip branch: out = x @ W + b   (plain fp32 WMMA GEMM, no norm/activation)
// One wave per 16x16 tile; block = 8 waves = 16 rows x 128 cols.
// A 16x4 frag: lane L -> row L&15, K = 2*(L>>4)+{0,1} (2 VGPRs)
// B 4x16 frag: lane L -> col L&15, same K split
// C 16x16:     VGPR v, lanes 0-15 -> row v, lanes 16-31 -> row v+8
// ---------------------------------------------------------------------------
__global__ __launch_bounds__(256)
void linear_kernel(const float* __restrict__ x, const float* __restrict__ W,
                   const float* __restrict__ b, float* __restrict__ out) {
    const int tid = threadIdx.x;
    const int wid = tid >> 5;
    const int lane = tid & 31;
    const int l16 = lane & 15;
    const int khalf = lane >> 4;
    const int m0 = blockIdx.x * 16;
    const int n0 = wid * 16;
    const int row = m0 + l16;
    const int col = n0 + l16;
    const float* xr = x + (size_t)row * D;

    v8f c = {};
    #pragma unroll 4
    for (int ks = 0; ks < D; ks += 4) {
        int ka = ks + 2 * khalf;
        v2f a, bb;
        a[0]  = xr[ka];            a[1]  = xr[ka + 1];
        bb[0] = W[ka * D + col];   bb[1] = W[(ka + 1) * D + col];
        c = __builtin_amdgcn_wmma_f32_16x16x4_f32(false, a, false, bb,
                                                  (short)0, c, false, false);
    }
    const float bias = b[col];
    #pragma unroll
    for (int v = 0; v < 8; ++v) {
        int rl = v + 8 * khalf;
        out[(size_t)(m0 + rl) * D + col] = c[v] + bias;
    }
}

// ---------------------------------------------------------------------------
// Fused SAGEConv tail:
//   o = (agg/deg) @ Wl + bl + hin @ Wr          (two fp32 WMMA chains)
//   o = o / max(||o||_2_row, eps)               (shfl butterfly + small LDS)
//   final_mode==0: out = prelu(o) + skip        (input of next layer)
//   final_mode==1: out = prelu(prelu(o))        (reference applies PReLU twice)
// ---------------------------------------------------------------------------
__global__ __launch_bounds__(256)
void sage_gemm_kernel(const float* __restrict__ agg, const float* __restrict__ deg,
                      const float* __restrict__ hin,
                      const float* __restrict__ Wl, const float* __restrict__ bl,
                      const float* __restrict__ Wr,
                      const float* __restrict__ skip,
                      const float* __restrict__ slope_p,
                      float* __restrict__ out, int final_mode) {
    __shared__ float rowsq[16];
    const int tid = threadIdx.x;
    const int wid = tid >> 5;
    const int lane = tid & 31;
    const int l16 = lane & 15;
    const int khalf = lane >> 4;
    const int m0 = blockIdx.x * 16;
    const int n0 = wid * 16;

    if (tid < 16) rowsq[tid] = 0.0f;
    __syncthreads();

    const int row = m0 + l16;
    const int col = n0 + l16;
    const float invd = 1.0f / fmaxf(deg[row], 1.0f);
    const float* aggRow = agg + (size_t)row * D;
    const float* hinRow = hin + (size_t)row * D;

    v8f c = {};
    #pragma unroll 4
    for (int ks = 0; ks < D; ks += 4) {
        int ka = ks + 2 * khalf;
        v2f aL, aR, bL, bR;
        aL[0] = aggRow[ka] * invd;  aL[1] = aggRow[ka + 1] * invd;
        aR[0] = hinRow[ka];         aR[1] = hinRow[ka + 1];
        bL[0] = Wl[ka * D + col];   bL[1] = Wl[(ka + 1) * D + col];
        bR[0] = Wr[ka * D + col];   bR[1] = Wr[(ka + 1) * D + col];
        c = __builtin_amdgcn_wmma_f32_16x16x4_f32(false, aL, false, bL,
                                                  (short)0, c, false, false);
        c = __builtin_amdgcn_wmma_f32_16x16x4_f32(false, aR, false, bR,
                                                  (short)0, c, false, false);
    }

    // bias + per-row sum of squares. XOR offsets 1,2,4,8 stay inside each
    // 16-lane half, matching the C layout (half 0 -> rows v, half 1 -> v+8).
    const float bias = bl[col];
    float vals[8];
    #pragma unroll
    for (int v = 0; v < 8; ++v) {
        float f = c[v] + bias;
        vals[v] = f;
        float s = f * f;
        s += __shfl_xor(s, 1, 32);
        s += __shfl_xor(s, 2, 32);
        s += __shfl_xor(s, 4, 32);
        s += __shfl_xor(s, 8, 32);
        if (l16 == 0) atomicAdd(&rowsq[v + 8 * khalf], s);  // 16 atomics/wave
    }
    __syncthreads();

    const float a = *slope_p;
    #pragma unroll
    for (int v = 0; v < 8; ++v) {
        int rl = v + 8 * khalf;
        float inv = 1.0f / fmaxf(sqrtf(rowsq[rl]), 1e-12f);
        float f = vals[v] * inv;
        f = (f >= 0.0f) ? f : a * f;               // PReLU
        size_t idx = (size_t)(m0 + rl) * D + col;
        if (final_mode) {
            f = (f >= 0.0f) ? f : a * f;           // second PReLU (reference)
            out[idx] = f;
        } else {
            out[idx] = f + skip[idx];
        }
    }
}

// ---------------------------------------------------------------------------
extern "C" void kernel_launch(void* const* d_in, const int* in_sizes, int n_in,
                              void* d_out, int out_size, void* d_ws, size_t ws_size,
                              hipStream_t stream) {
    const float*     x   = (const float*)d_in[0];
    const long long* ei  = (const long long*)d_in[1];   // int64 edge_index [2,E]
    const float* W1l = (const float*)d_in[2];
    const float* b1l = (const float*)d_in[3];
    const float* W1r = (const float*)d_in[4];
    const float* W2l = (const float*)d_in[5];
    const float* b2l = (const float*)d_in[6];
    const float* W2r = (const float*)d_in[7];
    const float* W3l = (const float*)d_in[8];
    const float* b3l = (const float*)d_in[9];
    const float* W3r = (const float*)d_in[10];
    const float* Ws  = (const float*)d_in[11];
    const float* bs  = (const float*)d_in[12];
    const float* a   = (const float*)d_in[13];          // PReLU slope (1 elem)

    float* ws    = (float*)d_ws;
    float* agg   = ws;
    float* hskip = ws + (size_t)NB;
    float* buf0  = ws + 2 * (size_t)NB;
    float* buf1  = ws + 3 * (size_t)NB;
    float* deg   = ws + 4 * (size_t)NB;
    float* out   = (float*)d_out;

    dim3 blk(256);
    dim3 gemm_grid(N_NODES / 16);                 // 3125 (exact)
    dim3 zero_grid((NB + 255) / 256);             // 25000
    dim3 scat_grid(N_EDGES / 256);                // 3125 blocks: 8 waves x 32 edges

    // skip branch: hskip = x @ Ws + bs
    linear_kernel<<<gemm_grid, blk, 0, stream>>>(x, Ws, bs, hskip);

    // layer 1: buf0 = prelu(sage(x)) + hskip
    zero_kernel<<<zero_grid, blk, 0, stream>>>(agg, deg);
    scatter_kernel<<<scat_grid, blk, 0, stream>>>(x, ei, agg, deg);
    sage_gemm_kernel<<<gemm_grid, blk, 0, stream>>>(agg, deg, x, W1l, b1l, W1r,
                                                    hskip, a, buf0, 0);
    // layer 2: buf1 = prelu(sage(buf0)) + hskip
    zero_kernel<<<zero_grid, blk, 0, stream>>>(agg, deg);
    scatter_kernel<<<scat_grid, blk, 0, stream>>>(buf0, ei, agg, deg);
    sage_gemm_kernel<<<gemm_grid, blk, 0, stream>>>(agg, deg, buf0, W2l, b2l, W2r,
                                                    hskip, a, buf1, 0);
    // layer 3: out = prelu(prelu(sage(buf1)))
    zero_kernel<<<zero_grid, blk, 0, stream>>>(agg, deg);
    scatter_kernel<<<scat_grid, blk, 0, stream>>>(buf1, ei, agg, deg);
    sage_gemm_kernel<<<gemm_grid, blk, 0, stream>>>(agg, deg, buf1, W3l, b3l, W3r,
                                                    nullptr, a, out, 1);
}